// DenseDilatedKnnGraph_7138235646515
// MI455X (gfx1250) — compile-verified
//
#include <hip/hip_runtime.h>

// ---------------------------------------------------------------------------
// DenseDilatedKnnGraph for MI455X (gfx1250, wave32).
// x: (B=2, C=64, N=8192, 1) fp32 -> edge_index (2, B, N, 9) int32
//
// Kernel 1: L2-normalize channels, transpose to (B, N, C) in workspace, and
//           store per-point squared norm of the normalized vector.
// Kernel 2: per workgroup (1 wave) handle 32 rows; scan all 8192 columns in
//           16-wide tiles using V_WMMA_F32_16X16X4_F32 (K=64 via 16 chained
//           K=4 WMMAs, two stacked M-tiles share the B fragment). Each lane
//           owns one row and keeps a sorted top-18 (dist, idx) list in LDS,
//           then emits every 2nd index (dilation) plus the center index.
// ---------------------------------------------------------------------------

typedef __attribute__((ext_vector_type(2))) float v2f;
typedef __attribute__((ext_vector_type(8))) float v8f;

#define N_PTS 8192
#define C_DIM 64
#define B_DIM 2
#define KD    18   // K * dilation
#define K_OUT 9

__global__ __launch_bounds__(256)
void knn_normalize_kernel(const float* __restrict__ x,
                          float* __restrict__ pts,
                          float* __restrict__ sq) {
  int p = blockIdx.x * blockDim.x + threadIdx.x;   // 0 .. B*N-1
  int b = p >> 13;                                  // N = 8192 = 2^13
  int n = p & (N_PTS - 1);
  const float* xb = x + (size_t)b * C_DIM * N_PTS + n;

  float v[C_DIM];
  float s = 0.0f;
#pragma unroll
  for (int c = 0; c < C_DIM; ++c) {
    v[c] = xb[(size_t)c * N_PTS];
    s += v[c] * v[c];
  }
  float m = fmaxf(sqrtf(s), 1e-12f);

  float s2 = 0.0f;
  float* dst = pts + (size_t)p * C_DIM;
#pragma unroll
  for (int c = 0; c < C_DIM; ++c) {
    float xn = v[c] / m;          // matches x / max(norm, eps)
    dst[c] = xn;
    s2 += xn * xn;
  }
  sq[p] = s2;
}

__global__ __launch_bounds__(32)
void knn_topk_kernel(const float* __restrict__ pts,
                     const float* __restrict__ sq,
                     int* __restrict__ out) {
  __shared__ float tile[32][17];     // -2*inner, padded to dodge bank conflicts
  __shared__ float sqJ[16];
  __shared__ float lDist[32][KD];
  __shared__ int   lIdx[32][KD];

  const int wg   = blockIdx.x;       // 0 .. B*(N/32)-1 = 0..511
  const int b    = wg >> 8;          // N/32 = 256 row-blocks per batch
  const int i0   = (wg & 255) * 32;
  const int lane = threadIdx.x;      // 0..31 (wave32)
  const int m16  = lane & 15;
  const int half = lane >> 4;

  const float* ptsB = pts + (size_t)b * N_PTS * C_DIM;
  const float* sqB  = sq  + (size_t)b * N_PTS;

  // --- preload A fragments for the two 16-row tiles (rows i0.. / i0+16..) ---
  // fp32 16x4 A layout: lane%16 = M, K = vgpr + 2*(lane>=16)
  v2f a0[16], a1[16];
#pragma unroll
  for (int kk = 0; kk < 16; ++kk) {
    int kbase = 4 * kk + 2 * half;
    a0[kk] = *(const v2f*)(ptsB + (size_t)(i0 + m16) * C_DIM + kbase);
    a1[kk] = *(const v2f*)(ptsB + (size_t)(i0 + 16 + m16) * C_DIM + kbase);
  }
  const float sqSelf = sqB[i0 + lane];   // row owned by this lane

  // --- init top-18 list (ascending distance) ---
#pragma unroll
  for (int e = 0; e < KD; ++e) {
    lDist[lane][e] = 3.0e38f;
    lIdx[lane][e]  = 0;
  }
  float worst = 3.0e38f;
  __syncthreads();

  // --- scan all columns in 16-wide tiles ---
  for (int j0 = 0; j0 < N_PTS; j0 += 16) {
    v8f c0 = {};
    v8f c1 = {};
#pragma unroll
    for (int kk = 0; kk < 16; ++kk) {
      int kbase = 4 * kk + 2 * half;
      // fp32 4x16 B layout mirrors A: lane%16 = N, K = vgpr + 2*(lane>=16)
      v2f bb = *(const v2f*)(ptsB + (size_t)(j0 + m16) * C_DIM + kbase);
      c0 = __builtin_amdgcn_wmma_f32_16x16x4_f32(false, a0[kk], false, bb,
                                                 (short)0, c0, false, false);
      c1 = __builtin_amdgcn_wmma_f32_16x16x4_f32(false, a1[kk], false, bb,
                                                 (short)0, c1, false, false);
    }

    if (lane < 16) sqJ[lane] = sqB[j0 + lane];
    // C/D layout: lane holds col = lane%16, rows r + 8*half (vgpr r)
#pragma unroll
    for (int r = 0; r < 8; ++r) {
      tile[half * 8 + r][m16]      = -2.0f * c0[r];
      tile[16 + half * 8 + r][m16] = -2.0f * c1[r];
    }
    __syncthreads();

    // selection: lane owns row `lane`; columns ascend -> stable tie-break
#pragma unroll 1
    for (int jj = 0; jj < 16; ++jj) {
      float d = (sqSelf + tile[lane][jj]) + sqJ[jj];  // (sq_i - 2*in) + sq_j
      if (d < worst) {
        int j = j0 + jj;
        int pos = KD - 1;
        while (pos > 0 && lDist[lane][pos - 1] > d) {
          lDist[lane][pos] = lDist[lane][pos - 1];
          lIdx[lane][pos]  = lIdx[lane][pos - 1];
          --pos;
        }
        lDist[lane][pos] = d;
        lIdx[lane][pos]  = j;
        worst = lDist[lane][KD - 1];
      }
    }
    __syncthreads();
  }

  // --- emit dilated neighbors: edge_index (2, B, N, 9) ---
  const int i = i0 + lane;
  int* out0 = out + ((size_t)b * N_PTS + i) * K_OUT;
  int* out1 = out0 + (size_t)B_DIM * N_PTS * K_OUT;
#pragma unroll
  for (int k = 0; k < K_OUT; ++k) {
    out0[k] = lIdx[lane][2 * k];   // nn_idx strided by dilation
    out1[k] = i;                   // center index
  }
}

extern "C" void kernel_launch(void* const* d_in, const int* in_sizes, int n_in,
                              void* d_out, int out_size, void* d_ws, size_t ws_size,
                              hipStream_t stream) {
  (void)in_sizes; (void)n_in; (void)out_size; (void)ws_size;
  const float* x = (const float*)d_in[0];

  float* pts = (float*)d_ws;                                  // B*N*C floats (4 MB)
  float* sq  = (float*)((char*)d_ws +
                        (size_t)B_DIM * N_PTS * C_DIM * sizeof(float)); // B*N floats
  int* out = (int*)d_out;                                     // (2,B,N,9) int32

  knn_normalize_kernel<<<(B_DIM * N_PTS) / 256, 256, 0, stream>>>(x, pts, sq);
  knn_topk_kernel<<<B_DIM * (N_PTS / 32), 32, 0, stream>>>(pts, sq, out);
}